// Chebnet_44238163149361
// MI455X (gfx1250) — compile-verified
//
#include <hip/hip_runtime.h>
#include <hip/hip_bf16.h>

typedef __attribute__((ext_vector_type(2))) float v2f;
typedef __attribute__((ext_vector_type(8))) float v8f;

#define BB   32
#define NN   1024
#define CC   64
#define EE   32768
#define MM   (BB * NN)          // 32768 rows
#define SZF  ((size_t)MM * CC)  // floats per activation tensor

// ---------------- elementwise helpers ----------------

__global__ __launch_bounds__(256) void k_zero(float* p, size_t n) {
  size_t i = (size_t)blockIdx.x * blockDim.x + threadIdx.x;
  size_t st = (size_t)gridDim.x * blockDim.x;
  for (; i < n; i += st) p[i] = 0.0f;
}

__global__ __launch_bounds__(256) void k_negcopy(float* __restrict__ d,
                                                 const float* __restrict__ s, size_t n) {
  size_t i = (size_t)blockIdx.x * blockDim.x + threadIdx.x;
  size_t st = (size_t)gridDim.x * blockDim.x;
  for (; i < n; i += st) d[i] = -s[i];
}

__global__ __launch_bounds__(256) void k_bias_relu(float* h, const float* __restrict__ bias,
                                                   size_t n, int C) {
  size_t i = (size_t)blockIdx.x * blockDim.x + threadIdx.x;
  size_t st = (size_t)gridDim.x * blockDim.x;
  for (; i < n; i += st) {
    int c = (int)(i % C);
    h[i] = fmaxf(h[i] + bias[c], 0.0f);
  }
}

// ---------------- graph normalization ----------------

__global__ __launch_bounds__(256) void k_deg(const int* __restrict__ src, float* deg, int E) {
  int e = blockIdx.x * blockDim.x + threadIdx.x;
  if (e < E) atomicAdd(&deg[src[e]], 1.0f);
}

__global__ __launch_bounds__(256) void k_norm(const int* __restrict__ src,
                                              const int* __restrict__ tgt,
                                              const float* __restrict__ deg,
                                              float* __restrict__ nrm, int E) {
  int e = blockIdx.x * blockDim.x + threadIdx.x;
  if (e >= E) return;
  float ds = deg[src[e]], dt = deg[tgt[e]];
  float a = ds > 0.0f ? rsqrtf(ds) : 0.0f;
  float b = dt > 0.0f ? rsqrtf(dt) : 0.0f;
  nrm[e] = -a * b;
}

// out[b,tgt,:] += scale * nrm[e] * x[b,src,:]   (block = 64 lanes = channels)
__global__ __launch_bounds__(64) void k_prop(float* __restrict__ out,
                                             const float* __restrict__ x,
                                             const int* __restrict__ src,
                                             const int* __restrict__ tgt,
                                             const float* __restrict__ nrm, float scale) {
  int e = blockIdx.x;
  int c = threadIdx.x;
  int s = src[e], t = tgt[e];
  float w = scale * nrm[e];
  for (int b = 0; b < BB; ++b) {
    float v = x[((size_t)b * NN + s) * CC + c] * w;
    atomicAdd(&out[((size_t)b * NN + t) * CC + c], v);
  }
}

// ---------------- WMMA f32 GEMM: H (+)= T[M,64] @ W[64,64] ----------------
// One wave -> one 16x16 output tile; K swept in steps of 4 (V_WMMA_F32_16X16X4_F32).

__global__ __launch_bounds__(256) void k_mm(float* __restrict__ H,
                                            const float* __restrict__ T,
                                            const float* __restrict__ W, int init) {
  int wave  = threadIdx.x >> 5;               // 0..7
  int lane  = threadIdx.x & 31;
  int otile = wave & 3;                       // 4 col tiles cover 64 cols
  int mtile = (blockIdx.x << 1) + (wave >> 2);
  int m0    = mtile * 16;
  int o0    = otile * 16;
  int half  = lane >> 4;                      // 0: lanes 0-15, 1: lanes 16-31
  int l15   = lane & 15;
  int row   = m0 + l15;                       // A-matrix row for this lane
  int ocol  = o0 + l15;                       // B/C/D column for this lane

  v8f acc;
  if (init) {
#pragma unroll
    for (int r = 0; r < 8; ++r) acc[r] = 0.0f;
  } else {
#pragma unroll
    for (int r = 0; r < 8; ++r)
      acc[r] = H[(size_t)(m0 + r + half * 8) * CC + ocol];
  }

#pragma unroll
  for (int c0 = 0; c0 < CC; c0 += 4) {
    int kbase = c0 + 2 * half;
    // A frag: lane holds (row, kbase) and (row, kbase+1)
    const float* ap = T + (size_t)row * CC + kbase;
    v2f a; a.x = ap[0]; a.y = ap[1];
    // B frag: lane holds (kbase, ocol) and (kbase+1, ocol)
    v2f b;
    b.x = W[(size_t)(kbase + 0) * CC + ocol];
    b.y = W[(size_t)(kbase + 1) * CC + ocol];
    acc = __builtin_amdgcn_wmma_f32_16x16x4_f32(false, a, false, b,
                                                (short)0, acc, false, false);
  }

#pragma unroll
  for (int r = 0; r < 8; ++r)
    H[(size_t)(m0 + r + half * 8) * CC + ocol] = acc[r];
}

// ---------------- BatchNorm over (B, F) per node, in place ----------------

__global__ __launch_bounds__(256) void k_bn(float* __restrict__ x,
                                            const float* __restrict__ g,
                                            const float* __restrict__ b,
                                            int F) {
  int n = blockIdx.x;
  int tid = threadIdx.x;
  int total = BB * F;
  __shared__ float s_sum[256];
  __shared__ float s_sq[256];
  float sum = 0.0f, sq = 0.0f;
  for (int i = tid; i < total; i += 256) {
    int bb = i / F, f = i % F;
    float v = x[((size_t)bb * NN + n) * F + f];
    sum += v; sq += v * v;
  }
  s_sum[tid] = sum; s_sq[tid] = sq;
  __syncthreads();
  for (int s = 128; s > 0; s >>= 1) {
    if (tid < s) { s_sum[tid] += s_sum[tid + s]; s_sq[tid] += s_sq[tid + s]; }
    __syncthreads();
  }
  float m   = s_sum[0] / (float)total;
  float var = s_sq[0] / (float)total - m * m;
  float inv = rsqrtf(var + 1e-5f);
  float scale = g[n] * inv;
  float shift = b[n] - m * scale;
  for (int i = tid; i < total; i += 256) {
    int bb = i / F, f = i % F;
    size_t idx = ((size_t)bb * NN + n) * F + f;
    x[idx] = x[idx] * scale + shift;
  }
}

// ---------------- layer 3 (C_out = 1, K = 3) + fused relu ----------------

__global__ __launch_bounds__(256) void k_cheb_out1(const float* __restrict__ T0,
                                                   const float* __restrict__ T1,
                                                   const float* __restrict__ T2,
                                                   const float* __restrict__ W,   // [3*64]
                                                   const float* __restrict__ bias,
                                                   float* __restrict__ out) {
  size_t i = (size_t)blockIdx.x * blockDim.x + threadIdx.x;
  if (i >= (size_t)MM) return;
  float acc = bias[0];
  const float* t0 = T0 + i * CC;
  const float* t1 = T1 + i * CC;
  const float* t2 = T2 + i * CC;
#pragma unroll 8
  for (int c = 0; c < CC; ++c)
    acc += t0[c] * W[c] + t1[c] * W[64 + c] + t2[c] * W[128 + c];
  out[i] = fmaxf(acc, 0.0f);
}

// ---------------- non-shared FCs ----------------

__global__ __launch_bounds__(256) void k_nsfc1(const float* __restrict__ h1,   // [B*N]
                                               const float* __restrict__ w,    // [N,16]
                                               const float* __restrict__ bb,   // [N,16]
                                               float* __restrict__ h16) {
  size_t idx = (size_t)blockIdx.x * blockDim.x + threadIdx.x;  // B*N*16
  if (idx >= (size_t)MM * 16) return;
  int j  = (int)(idx & 15);
  size_t bn = idx >> 4;
  int n = (int)(bn % NN);
  float v = h1[bn] * w[(size_t)n * 16 + j] + bb[(size_t)n * 16 + j];
  h16[idx] = fmaxf(v, 0.0f);  // relu (applied before final BN)
}

__global__ __launch_bounds__(256) void k_nsfc2(const float* __restrict__ h16,  // [B*N,16]
                                               const float* __restrict__ w,    // [N,16]
                                               const float* __restrict__ bb,   // [N,1]
                                               float* __restrict__ out) {
  size_t bn = (size_t)blockIdx.x * blockDim.x + threadIdx.x;
  if (bn >= (size_t)MM) return;
  int n = (int)(bn % NN);
  float acc = bb[n];
#pragma unroll
  for (int j = 0; j < 16; ++j)
    acc += h16[bn * 16 + j] * w[(size_t)n * 16 + j];
  out[bn] = acc;
}

// ---------------- host-side layer driver ----------------

static void cheb_layer(const float* X, const float* Wk, const float* bias,
                       const float* bn_g, const float* bn_b,
                       float* H, float* Ta, float* Tb, float* Tc,
                       const int* src, const int* tgt, const float* nrm,
                       int K, hipStream_t stream) {
  const int mmGrid = MM / 32;          // 2 row-tiles of 16 per block
  const int ewGrid = (int)(SZF / 256); // elementwise grid

  // k = 0
  k_mm<<<mmGrid, 256, 0, stream>>>(H, X, Wk, 1);

  const float* T_km2 = X;
  const float* T_km1 = nullptr;
  if (K > 1) {  // k = 1: T1 = prop(X)
    k_zero<<<ewGrid, 256, 0, stream>>>(Ta, SZF);
    k_prop<<<EE, 64, 0, stream>>>(Ta, X, src, tgt, nrm, 1.0f);
    k_mm<<<mmGrid, 256, 0, stream>>>(H, Ta, Wk + 64 * 64, 0);
    T_km1 = Ta;
  }
  float* rot[3] = {Tc, Ta, Tb};        // k=1->Ta, k=2->Tb, k=3->Tc, k=4->Ta
  for (int k = 2; k < K; ++k) {
    float* Tk = rot[k % 3];
    k_negcopy<<<ewGrid, 256, 0, stream>>>(Tk, T_km2, SZF);       // Tk = -T_{k-2}
    k_prop<<<EE, 64, 0, stream>>>(Tk, T_km1, src, tgt, nrm, 2.0f); // += 2*prop(T_{k-1})
    k_mm<<<mmGrid, 256, 0, stream>>>(H, Tk, Wk + (size_t)k * 64 * 64, 0);
    T_km2 = T_km1; T_km1 = Tk;
  }
  k_bias_relu<<<ewGrid, 256, 0, stream>>>(H, bias, SZF, CC);
  k_bn<<<NN, 256, 0, stream>>>(H, bn_g, bn_b, CC);
}

extern "C" void kernel_launch(void* const* d_in, const int* in_sizes, int n_in,
                              void* d_out, int out_size, void* d_ws, size_t ws_size,
                              hipStream_t stream) {
  const float* x       = (const float*)d_in[0];
  const int*   edge    = (const int*)d_in[1];
  const float* cheb_w1 = (const float*)d_in[2];
  const float* cheb_b1 = (const float*)d_in[3];
  const float* cheb_w2 = (const float*)d_in[4];
  const float* cheb_b2 = (const float*)d_in[5];
  const float* cheb_w3 = (const float*)d_in[6];
  const float* cheb_b3 = (const float*)d_in[7];
  const float* bn_g1 = (const float*)d_in[8];
  const float* bn_b1 = (const float*)d_in[9];
  const float* bn_g2 = (const float*)d_in[10];
  const float* bn_b2 = (const float*)d_in[11];
  const float* bn_g3 = (const float*)d_in[12];
  const float* bn_b3 = (const float*)d_in[13];
  const float* bn_g4 = (const float*)d_in[14];
  const float* bn_b4 = (const float*)d_in[15];
  const float* fc_w1 = (const float*)d_in[16];
  const float* fc_b1 = (const float*)d_in[17];
  const float* fc_w2 = (const float*)d_in[18];
  const float* fc_b2 = (const float*)d_in[19];

  const int* src = edge;        // edge_index[0]
  const int* tgt = edge + EE;   // edge_index[1]

  // workspace carve-up (floats)
  float* ws  = (float*)d_ws;
  float* H1  = ws;                 // [M,64]
  float* H2  = H1 + SZF;           // [M,64]
  float* Ta  = H2 + SZF;           // [M,64]
  float* Tb  = Ta + SZF;           // [M,64]
  float* Tc  = Tb + SZF;           // [M,64]
  float* deg = Tc + SZF;           // [N]
  float* nrm = deg + NN;           // [E]
  float* h1b = nrm + EE;           // [M]
  float* h16 = h1b + MM;           // [M,16]

  // ---- graph normalization ----
  k_zero<<<(NN + 255) / 256, 256, 0, stream>>>(deg, NN);
  k_deg<<<EE / 256, 256, 0, stream>>>(src, deg, EE);
  k_norm<<<EE / 256, 256, 0, stream>>>(src, tgt, deg, nrm, EE);

  // ---- ChebConv 1 + relu + BN ----
  cheb_layer(x,  cheb_w1, cheb_b1, bn_g1, bn_b1, H1, Ta, Tb, Tc, src, tgt, nrm, 5, stream);
  // ---- ChebConv 2 + relu + BN ----
  cheb_layer(H1, cheb_w2, cheb_b2, bn_g2, bn_b2, H2, Ta, Tb, Tc, src, tgt, nrm, 5, stream);

  // ---- ChebConv 3 (K=3, C_out=1) + relu ----
  const int ewGrid = (int)(SZF / 256);
  k_zero<<<ewGrid, 256, 0, stream>>>(Ta, SZF);
  k_prop<<<EE, 64, 0, stream>>>(Ta, H2, src, tgt, nrm, 1.0f);     // T1
  k_negcopy<<<ewGrid, 256, 0, stream>>>(Tb, H2, SZF);             // T2 = -T0
  k_prop<<<EE, 64, 0, stream>>>(Tb, Ta, src, tgt, nrm, 2.0f);     //    + 2*prop(T1)
  k_cheb_out1<<<MM / 256, 256, 0, stream>>>(H2, Ta, Tb, cheb_w3, cheb_b3, h1b);
  k_bn<<<NN, 256, 0, stream>>>(h1b, bn_g3, bn_b3, 1);

  // ---- NSFC1 + relu + BN ----
  k_nsfc1<<<(MM * 16) / 256, 256, 0, stream>>>(h1b, fc_w1, fc_b1, h16);
  k_bn<<<NN, 256, 0, stream>>>(h16, bn_g4, bn_b4, 16);

  // ---- NSFC2 -> output [B,N] ----
  k_nsfc2<<<MM / 256, 256, 0, stream>>>(h16, fc_w2, fc_b2, (float*)d_out);
}